// Net_9887014715533
// MI455X (gfx1250) — compile-verified
//
#include <hip/hip_runtime.h>
#include <math.h>

// cosine similarity of x[4096] against memory[65536][4096] on gfx1250.
// Bandwidth-bound (1.07 GB of memory traffic, ~46us floor at 23.3 TB/s).
// Dot products done with V_WMMA_F32_16X16X4_F32: B's 16 columns all hold the
// same x K-chunk, so every column of D is the row dot product. K-order inside
// each WMMA is permuted to match a single b128 load per lane (no shuffles).

typedef float v8f __attribute__((ext_vector_type(8)));
typedef float v2f __attribute__((ext_vector_type(2)));

#define INFEAT 4096
#define CAPACITY 65536
#define EPS 1e-8f

__global__ __launch_bounds__(256, 2)
void cosine_gemv_wmma(const float* __restrict__ x,
                      const float* __restrict__ memory,
                      float* __restrict__ out) {
    __shared__ float xs[INFEAT];       // 16 KB copy of x
    __shared__ float dots[8 * 16];     // per-wave row dot products
    __shared__ float xnorm_sq;

    const int tid  = threadIdx.x;
    const int lane = tid & 31;
    const int wave = tid >> 5;

    if (tid == 0) xnorm_sq = 0.0f;
    __syncthreads();

    // Cooperative load of x into LDS (float4 per thread x4) + ||x||^2 partial.
    float xss = 0.0f;
#pragma unroll
    for (int i = 0; i < INFEAT / (256 * 4); ++i) {
        int idx = (i * 256 + tid) * 4;
        float4 v = *(const float4*)(x + idx);
        *(float4*)(xs + idx) = v;
        xss += v.x * v.x + v.y * v.y + v.z * v.z + v.w * v.w;
    }
    atomicAdd(&xnorm_sq, xss);         // ds_add_f32
    __syncthreads();
    const float xnorm = sqrtf(xnorm_sq);

    // Each wave owns 16 rows. Lane L and L+16 both read row (L&15):
    // lane<16 reads K[k0..k0+3], lane>=16 reads K[k0+4..k0+7] (one b128 each).
    const int row_base = blockIdx.x * 128 + wave * 16;
    const int row      = row_base + (lane & 15);
    const int colsel   = (lane >> 4) << 2;   // 0 or 4 floats

    const float* __restrict__ aptr = memory + (size_t)row * INFEAT + colsel;
    const float* __restrict__ bptr = xs + colsel;

    v8f  c  = {};       // D accumulator: 16x16 f32, all columns identical
    float ss = 0.0f;    // per-lane partial sum of squares of this row

#pragma unroll 4
    for (int k0 = 0; k0 < INFEAT; k0 += 8) {
        float4 av = *(const float4*)(aptr + k0);   // global_load_b128
        float4 bv = *(const float4*)(bptr + k0);   // ds_load_b128 (broadcast)

        v2f a1 = {av.x, av.y}, a2 = {av.z, av.w};
        v2f b1 = {bv.x, bv.y}, b2 = {bv.z, bv.w};

        // Two K=4 WMMAs per 8 K values; K permutation identical in A and B.
        c = __builtin_amdgcn_wmma_f32_16x16x4_f32(false, a1, false, b1,
                                                  (short)0, c, false, false);
        c = __builtin_amdgcn_wmma_f32_16x16x4_f32(false, a2, false, b2,
                                                  (short)0, c, false, false);

        ss += av.x * av.x + av.y * av.y + av.z * av.z + av.w * av.w;
    }

    // Fold lane pairs (L, L+16): lane L now holds full sumsq of row (L&15).
    ss += __shfl_xor(ss, 16, 32);

    // D column N=0: lane 0 holds rows 0..7 (VGPR 0..7), lane 16 rows 8..15.
    if ((lane & 15) == 0) {
        float* dst = dots + wave * 16 + (lane >> 4) * 8;
#pragma unroll
        for (int r = 0; r < 8; ++r) dst[r] = c[r];
    }
    __syncthreads();

    if (lane < 16) {
        float dot   = dots[wave * 16 + lane];
        float denom = fmaxf(sqrtf(ss) * xnorm, EPS);
        out[row] = dot / denom;
    }
}

extern "C" void kernel_launch(void* const* d_in, const int* in_sizes, int n_in,
                              void* d_out, int out_size, void* d_ws, size_t ws_size,
                              hipStream_t stream) {
    const float* x      = (const float*)d_in[0];   // [4096]
    const float* memory = (const float*)d_in[1];   // [65536, 4096]
    float*       out    = (float*)d_out;           // [65536]

    dim3 grid(CAPACITY / 128);   // 8 waves * 16 rows per block
    dim3 block(256);
    hipLaunchKernelGGL(cosine_gemv_wmma, grid, block, 0, stream, x, memory, out);
}